// TopKActivation_80582176408300
// MI455X (gfx1250) — compile-verified
//
#include <hip/hip_runtime.h>
#include <stdint.h>

// TopK activation, MI455X (gfx1250).
// Memory-bound: 512 MB total traffic @ 23.3 TB/s => ~22 us floor.
// One 256-thread block per row; async-DMA the 64KB row into LDS
// (CDNA5 GLOBAL_LOAD_ASYNC_TO_LDS_B128 + s_wait_asynccnt), radix-select the
// K-th largest value entirely in LDS (320KB/WGP makes the whole row fit),
// stream the masked row back out NT. HBM sees exactly 1 read + 1 write/elem.

#define NROW_ELEMS 16384
#define TOPK       100
#define NTHREADS   256
#define LISTCAP    2048   // candidate-key list after round 0 (expected ~a few hundred)

typedef float  fv4 __attribute__((ext_vector_type(4)));
typedef unsigned uv4 __attribute__((ext_vector_type(4)));

// Monotonic float->uint key: preserves float ordering. 2 VALU ops.
__device__ __forceinline__ unsigned toKey(unsigned u) {
  return u ^ (unsigned)(((int)u >> 31) | (int)0x80000000);
}
__device__ __forceinline__ unsigned fromKey(unsigned k) {
  return (k & 0x80000000u) ? (k ^ 0x80000000u) : ~k;
}

__global__ __launch_bounds__(NTHREADS)
void topk_row_select(const float* __restrict__ x, float* __restrict__ out, int ncols) {
  extern __shared__ __align__(16) unsigned smem[];
  unsigned* vals = smem;                     // ncols raw float bit patterns
  unsigned* hist = smem + NROW_ELEMS;        // 256 bins
  unsigned* list = hist + 256;               // LISTCAP candidate keys
  unsigned* misc = list + LISTCAP;           // [0]=prefix [1]=kRem [2]=c_eq [3]=list count
  unsigned* wsum = misc + 8;                 // 8 per-wave partial sums

  const int tid  = threadIdx.x;
  const int lane = tid & 31;                 // wave32
  const int wav  = tid >> 5;                 // 8 waves
  const size_t rbase = (size_t)blockIdx.x * (size_t)ncols;

  if (tid == 0) misc[3] = 0;

  // ---------------- stage row into LDS (async 128b per lane) ----------------
#if __has_builtin(__builtin_amdgcn_global_load_async_to_lds_b128)
  {
    typedef int v4i __attribute__((vector_size(16)));
    typedef __attribute__((address_space(1))) v4i gv4i;
    typedef __attribute__((address_space(3))) v4i lv4i;
    gv4i* g = (gv4i*)(uintptr_t)(x + rbase);
    lv4i* l = (lv4i*)(uint32_t)(uintptr_t)vals;  // generic LDS addr[31:0] == LDS offset
    #pragma unroll
    for (int i = 0; i < NROW_ELEMS / (NTHREADS * 4); ++i) {
      int v = i * NTHREADS + tid;
      __builtin_amdgcn_global_load_async_to_lds_b128(g + v, l + v, 0, 0);
    }
#if __has_builtin(__builtin_amdgcn_s_wait_asynccnt)
    __builtin_amdgcn_s_wait_asynccnt(0);
#else
    asm volatile("s_wait_asynccnt 0" ::: "memory");
#endif
  }
#else
  {
    const fv4* gs = (const fv4*)(x + rbase);
    fv4* ls = (fv4*)vals;
    #pragma unroll
    for (int i = 0; i < NROW_ELEMS / (NTHREADS * 4); ++i) {
      int v = i * NTHREADS + tid;
      ls[v] = gs[v];
    }
  }
#endif
  __syncthreads();

  // ---------------- radix-select the K-th largest key ----------------
  unsigned prefix = 0, mask = 0, kRem = TOPK, ceq = 0;

  for (int r = 0; r < 4; ++r) {
    const int shift = 24 - 8 * r;

    hist[tid] = 0;
    __syncthreads();

    const unsigned lcnt = (r > 0) ? misc[3] : 0u;
    const bool useList = (r > 0) && (lcnt <= LISTCAP);

    if (useList) {
      for (unsigned j = tid; j < lcnt; j += NTHREADS) {
        unsigned key = list[j];
        if ((key & mask) == prefix) atomicAdd(&hist[(key >> shift) & 255u], 1u);
      }
    } else {
      for (int j = tid; j < NROW_ELEMS; j += NTHREADS) {
        unsigned key = toKey(vals[j]);
        if ((key & mask) == prefix) atomicAdd(&hist[(key >> shift) & 255u], 1u);
      }
    }
    __syncthreads();

    // Suffix sum hist[d] := #elements with digit >= d.
    // Intra-wave shuffle scan (wave32, no barriers) + cross-wave fixup.
    {
      unsigned v = hist[tid];
      #pragma unroll
      for (int off = 1; off < 32; off <<= 1) {
        unsigned o = __shfl_down(v, off, 32);
        if (lane + off < 32) v += o;
      }
      if (lane == 0) wsum[wav] = v;   // wave's 32-bin total
      __syncthreads();
      unsigned add = 0;
      #pragma unroll
      for (int w = 0; w < 8; ++w) add += (w > wav) ? wsum[w] : 0u;
      // v = suffix within wave starting at this lane; fold in higher waves
      hist[tid] = v + add;
      __syncthreads();
    }

    // Exactly one digit d satisfies sfx(d) >= kRem > sfx(d+1)
    {
      unsigned sd  = hist[tid];
      unsigned sd1 = (tid < 255) ? hist[tid + 1] : 0u;
      if (sd >= kRem && sd1 < kRem) {
        misc[0] = prefix | ((unsigned)tid << shift);
        misc[1] = kRem - sd1;   // rank remaining inside chosen digit
        misc[2] = sd - sd1;     // elements equal to chosen digit (final round: c_eq)
      }
    }
    __syncthreads();
    prefix = misc[0];
    kRem   = misc[1];
    ceq    = misc[2];
    mask  |= 0xFFu << shift;
    __syncthreads();

    // After round 0: compact candidates sharing the chosen top byte so
    // rounds 1-3 scan ~1/256 of the data.
    if (r == 0) {
      for (int j = tid; j < NROW_ELEMS; j += NTHREADS) {
        unsigned key = toKey(vals[j]);
        if ((key & mask) == prefix) {
          unsigned slot = atomicAdd(&misc[3], 1u);
          if (slot < LISTCAP) list[slot] = key;
        }
      }
      __syncthreads();
    }
  }

  const unsigned Tkey = prefix;   // exact key of the K-th largest element
  const unsigned need = kRem;     // how many elements == Tkey must be kept

  // Rare exact-tie path: reference (jax top_k) keeps lowest indices among
  // equal values. Condition is block-uniform.
  if (ceq != need) {
    if (tid == 0) {
      unsigned drop = fromKey(Tkey - 1u);  // any bit pattern with key < Tkey
      unsigned taken = 0;
      for (int j = 0; j < NROW_ELEMS; ++j) {
        if (toKey(vals[j]) == Tkey) {
          if (taken < need) ++taken;
          else vals[j] = drop;
        }
      }
    }
    __syncthreads();
  }

  // ---------------- masked write-back (NT 128b stores) ----------------
  {
    fv4* go = (fv4*)(out + rbase);
    const uv4* lv = (const uv4*)vals;
    #pragma unroll
    for (int i = 0; i < NROW_ELEMS / (NTHREADS * 4); ++i) {
      int v = i * NTHREADS + tid;
      uv4 u = lv[v];
      fv4 f;
      f.x = (toKey(u.x) >= Tkey) ? __uint_as_float(u.x) : 0.0f;
      f.y = (toKey(u.y) >= Tkey) ? __uint_as_float(u.y) : 0.0f;
      f.z = (toKey(u.z) >= Tkey) ? __uint_as_float(u.z) : 0.0f;
      f.w = (toKey(u.w) >= Tkey) ? __uint_as_float(u.w) : 0.0f;
      __builtin_nontemporal_store(f, go + v);   // touch-once stream, keep L2 clean
    }
  }
}

extern "C" void kernel_launch(void* const* d_in, const int* in_sizes, int n_in,
                              void* d_out, int out_size, void* d_ws, size_t ws_size,
                              hipStream_t stream) {
  (void)n_in; (void)d_ws; (void)ws_size; (void)out_size;
  const float* x = (const float*)d_in[0];
  float* out = (float*)d_out;

  const int ncols = NROW_ELEMS;                 // 16384 per reference
  const int nrows = in_sizes[0] / ncols;        // 4096 per reference

  const size_t shmem =
      (size_t)(NROW_ELEMS + 256 + LISTCAP + 16) * sizeof(unsigned);  // ~73 KB (<320 KB/WGP)

  // Opt in to >64KB dynamic LDS (no-op / ignored where unnecessary).
  (void)hipFuncSetAttribute((const void*)topk_row_select,
                            hipFuncAttributeMaxDynamicSharedMemorySize, (int)shmem);

  topk_row_select<<<dim3(nrows), dim3(NTHREADS), shmem, stream>>>(x, out, ncols);
}